// PhysicsConsistency_63797444215001
// MI455X (gfx1250) — compile-verified
//
#include <hip/hip_runtime.h>
#include <hip/hip_bf16.h>
#include <stdint.h>

// Problem constants (from reference: x = (256, 8192, 16) fp32, out = (256, 8191, 20) fp32)
#define BATCH   256
#define SEQ     8192
#define CIN     16
#define COUT    20
#define WINDOW  20
#define TILE    256                   // output timesteps per block
#define HALO    20                    // lookback rows needed (kd window + jerk)
#define ROWS    (TILE + HALO + 1)     // 277 input rows staged per tile
#define LSTRIDE 20                    // padded LDS row stride in floats (80B, 16B-aligned chunks)
#define INV_DT   200.0f               // 1/0.005
#define INV_DT2  40000.0f             // 1/0.005^2
#define DT_F     0.005f

__device__ __forceinline__ float nrm3(float a, float b, float c) {
    float s = a * a + b * b + c * c;
    return (s > 0.0f) ? sqrtf(s) : 0.0f;
}

__global__ __launch_bounds__(256) void
PhysicsConsistency_63797444215001_kernel(const float* __restrict__ x,
                                         float* __restrict__ out)
{
    __shared__ __align__(16) float smem[ROWS * LSTRIDE];

    const int tid = threadIdx.x;
    const int t0  = blockIdx.x * TILE;        // first output timestep of this tile
    const int b   = blockIdx.y;

    // ---------------------------------------------------------------
    // Stage input rows [t0-HALO, t0+TILE] into LDS with the CDNA5
    // async DMA path: global_load_async_to_lds_b128 (ASYNCcnt).
    // Per-lane LDS addresses let us pad the LDS row stride to 20
    // floats (bank-conflict mitigation) while global reads stay
    // fully coalesced 16B chunks.
    // ---------------------------------------------------------------
    const uint32_t bRow  = (uint32_t)b * (uint32_t)SEQ;
    const uint64_t xbase = (uint64_t)(uintptr_t)x;

    for (int e = tid; e < ROWS * 4; e += 256) {
        const int r = e >> 2;                 // staged row index
        const int c = e & 3;                  // 16B chunk within row
        const int g = t0 - HALO + r;          // global timestep of this row
        if (g >= 0 && g < SEQ) {
            const uint32_t goff  = ((bRow + (uint32_t)g) << 6) + ((uint32_t)c << 4);
            const uint32_t lbyte = (uint32_t)(uintptr_t)&smem[r * LSTRIDE + c * 4];
            asm volatile("global_load_async_to_lds_b128 %0, %1, %2"
                         :: "v"(lbyte), "v"(goff), "s"(xbase)
                         : "memory");
        }
    }
    asm volatile("s_wait_asynccnt 0" ::: "memory");
    __syncthreads();

    // ---------------------------------------------------------------
    // One output row (20 floats) per thread, all operands from LDS.
    // ---------------------------------------------------------------
    const int t = t0 + tid;
    if (t >= SEQ - 1) return;

    const float* rw = &smem[(tid + HALO) * LSTRIDE];  // row t
    const float* rp = rw + LSTRIDE;                   // row t+1
    const float* rm = rw - LSTRIDE;                   // row t-1

    // channels: pos=0..2, att=3..5, rate=6..8, vel=9..11
    // pos_vel_diff = (pos[t+1]-pos[t])/DT - vel[t+1]
    const float pvx = (rp[0] - rw[0]) * INV_DT - rp[9];
    const float pvy = (rp[1] - rw[1]) * INV_DT - rp[10];
    const float pvz = (rp[2] - rw[2]) * INV_DT - rp[11];

    // att_rate_diff = (att[t+1]-att[t])/DT - rate[t+1]
    const float arx = (rp[3] - rw[3]) * INV_DT - rp[6];
    const float ary = (rp[4] - rw[4]) * INV_DT - rp[7];
    const float arz = (rp[5] - rw[5]) * INV_DT - rp[8];

    // jerk_padded: 0 at t==0, else (vel[t+1]-2*vel[t]+vel[t-1])/DT^2
    float jx = 0.0f, jy = 0.0f, jz = 0.0f;
    if (t >= 1) {
        jx = (rp[9]  - 2.0f * rw[9]  + rm[9])  * INV_DT2;
        jy = (rp[10] - 2.0f * rw[10] + rm[10]) * INV_DT2;
        jz = (rp[11] - 2.0f * rw[11] + rm[11]) * INV_DT2;
    }

    // angular_accel = (rate[t+1]-rate[t])/DT
    const float aax = (rp[6] - rw[6]) * INV_DT;
    const float aay = (rp[7] - rw[7]) * INV_DT;
    const float aaz = (rp[8] - rw[8]) * INV_DT;

    // kd: for t>=20: DT*sum_{k=t-19..t} vel[k] - (pos[t+1]-pos[t-19]); else 0
    float kdx = 0.0f, kdy = 0.0f, kdz = 0.0f;
    if (t >= WINDOW) {
        const float* w = &smem[(tid + 1) * LSTRIDE];  // row t-19
        float sx = 0.0f, sy = 0.0f, sz = 0.0f;
        #pragma unroll
        for (int k = 0; k < WINDOW; ++k) {
            sx += w[9]; sy += w[10]; sz += w[11];
            w += LSTRIDE;
        }
        const float* r19 = &smem[(tid + 1) * LSTRIDE]; // pos at t-19
        kdx = DT_F * sx - (rp[0] - r19[0]);
        kdy = DT_F * sy - (rp[1] - r19[1]);
        kdz = DT_F * sz - (rp[2] - r19[2]);
    }

    __align__(16) float o[COUT];
    o[0]  = pvx; o[1]  = pvy; o[2]  = pvz; o[3]  = nrm3(pvx, pvy, pvz);
    o[4]  = arx; o[5]  = ary; o[6]  = arz; o[7]  = nrm3(arx, ary, arz);
    o[8]  = jx;  o[9]  = jy;  o[10] = jz;  o[11] = nrm3(jx, jy, jz);
    o[12] = aax; o[13] = aay; o[14] = aaz; o[15] = nrm3(aax, aay, aaz);
    o[16] = kdx; o[17] = kdy; o[18] = kdz; o[19] = nrm3(kdx, kdy, kdz);

    float4* op = (float4*)(out + ((size_t)b * (SEQ - 1) + (size_t)t) * COUT);
    const float4* ov = (const float4*)o;
    op[0] = ov[0];
    op[1] = ov[1];
    op[2] = ov[2];
    op[3] = ov[3];
    op[4] = ov[4];
}

extern "C" void kernel_launch(void* const* d_in, const int* in_sizes, int n_in,
                              void* d_out, int out_size, void* d_ws, size_t ws_size,
                              hipStream_t stream) {
    (void)in_sizes; (void)n_in; (void)d_ws; (void)ws_size; (void)out_size;
    const float* x = (const float*)d_in[0];
    float* out = (float*)d_out;
    dim3 grid((SEQ - 1 + TILE - 1) / TILE, BATCH, 1);   // 32 x 256 blocks
    PhysicsConsistency_63797444215001_kernel<<<grid, 256, 0, stream>>>(x, out);
}